// BilinearAttnPool_5755256176722
// MI455X (gfx1250) — compile-verified
//
#include <hip/hip_runtime.h>
#include <math.h>

typedef __attribute__((ext_vector_type(2))) float v2f;
typedef __attribute__((ext_vector_type(8))) float v8f;

#define C_DIM 2048
#define HW_DIM 288
#define M_DIM 8
#define B_DIM 32
#define MC_DIM (M_DIM * C_DIM)   // 16384
#define CTILES (C_DIM / 16)      // 128
#define K1_WAVES 4               // K-split waves in kernel 1
#define K1_CHUNK (C_DIM / K1_WAVES)  // 512

// ---------------------------------------------------------------------------
// Kernel 0: zero-pad conv_w (8x2048) into Wpad (16x2048). Rows 8..15 = 0, so
// kernel 1's A-operand loads need no per-iteration masking around the WMMA.
// ---------------------------------------------------------------------------
__global__ __launch_bounds__(256)
void pad_w_kernel(const float* __restrict__ W, float* __restrict__ Wpad) {
    const int i = blockIdx.x * 256 + threadIdx.x;   // 0 .. 16*2048-1
    const int m = i >> 11;                           // /2048
    Wpad[i] = (m < M_DIM) ? W[i] : 0.0f;
}

// ---------------------------------------------------------------------------
// Kernel 1: attn³[b,m,hw] = relu(BN(Σ_c F[b,c,hw]·Wpad[m,c]))³ via f32 WMMA.
// Block = 4 waves, each covering a K-chunk of 512; deterministic LDS tree
// reduction of the four 16x16 f32 accumulator tiles, then BN/ReLU/cube.
// A = Wpad (16m x 4c), B = F (4c x 16hw). Output padded to 16 m-rows.
// ---------------------------------------------------------------------------
__global__ __launch_bounds__(128)
void attn_cube_kernel(const float* __restrict__ F, const float* __restrict__ Wpad,
                      const float* __restrict__ bnS, const float* __restrict__ bnB,
                      const float* __restrict__ bnM, const float* __restrict__ bnV,
                      float* __restrict__ A3) {
    const int b     = blockIdx.y;
    const int hw0   = blockIdx.x * 16;
    const int wave  = threadIdx.x >> 5;
    const int lane  = threadIdx.x & 31;
    const int l15   = lane & 15;
    const int khalf = (lane >> 4) << 1;          // 0 or 2

    v8f acc = {0.f, 0.f, 0.f, 0.f, 0.f, 0.f, 0.f, 0.f};

    const float* fcol = F + (size_t)b * C_DIM * HW_DIM + hw0 + l15;
    const float* wrow = Wpad + (size_t)l15 * C_DIM + khalf;

    const int kend = wave * K1_CHUNK + K1_CHUNK;
    for (int k = wave * K1_CHUNK; k < kend; k += 4) {
        v2f a = *(const v2f*)(wrow + k);                     // Wpad, 8B aligned
        v2f bv;
        bv.x = fcol[(size_t)(k + khalf)     * HW_DIM];       // F, strided
        bv.y = fcol[(size_t)(k + khalf + 1) * HW_DIM];
        acc = __builtin_amdgcn_wmma_f32_16x16x4_f32(
            false, a, false, bv, (short)0, acc, false, false);
    }

    // Deterministic cross-wave reduction of the 16x16 tile.
    __shared__ float red[K1_WAVES][256];
#pragma unroll
    for (int r = 0; r < 8; ++r) red[wave][r * 32 + lane] = acc[r];
    __syncthreads();

    if (wave == 0) {
        const int mbase = (lane >> 4) << 3;      // D: m = r + (lane>=16 ? 8:0)
        float* orow = A3 + (size_t)b * 16 * HW_DIM + hw0 + l15;
#pragma unroll
        for (int r = 0; r < 8; ++r) {
            const int idx = r * 32 + lane;
            const float s = ((red[0][idx] + red[1][idx]) +
                             (red[2][idx] + red[3][idx]));
            const int m = r + mbase;
            float val = 0.0f;
            if (m < M_DIM) {
                const float inv = rsqrtf(bnV[m] + 1e-3f);
                float x = (s - bnM[m]) * (bnS[m] * inv) + bnB[m];
                x = fmaxf(x, 0.0f);
                val = x * x * x;                 // attn³
            }
            orow[(size_t)m * HW_DIM] = val;      // rows m=8..15 stay zero
        }
    }
}

// ---------------------------------------------------------------------------
// Kernel 2: P[b,c,m] = Σ_hw relu(F)³ · attn³  via f32 WMMA, K = 288.
// 8 waves/block, one 16c x 16m tile per wave (F is L2-resident by now).
// Epilogue: cbrt(mean), sign-sqrt, scattered store + per-wave partial Σ val².
// ---------------------------------------------------------------------------
__global__ __launch_bounds__(256)
void gem_pool_kernel(const float* __restrict__ F, const float* __restrict__ A3,
                     float* __restrict__ out, float* __restrict__ partial) {
    const int b     = blockIdx.y;
    const int wave  = threadIdx.x >> 5;
    const int lane  = threadIdx.x & 31;
    const int ctile = blockIdx.x * 8 + wave;
    const int c0    = ctile * 16;
    const int l15   = lane & 15;
    const int khalf = (lane >> 4) << 1;

    v8f acc = {0.f, 0.f, 0.f, 0.f, 0.f, 0.f, 0.f, 0.f};

    const float* frow = F  + ((size_t)b * C_DIM + c0 + l15) * HW_DIM + khalf;
    const float* arow = A3 + ((size_t)b * 16 + l15) * HW_DIM + khalf;

    for (int hw = 0; hw < HW_DIM; hw += 4) {
        v2f f2 = *(const v2f*)(frow + hw);                   // F, contiguous 8B
        v2f a;
        float fx = fmaxf(f2.x, 0.0f); a.x = fx * fx * fx;    // relu(F)³ fused
        float fy = fmaxf(f2.y, 0.0f); a.y = fy * fy * fy;
        v2f bv = *(const v2f*)(arow + hw);                   // attn³ (padded)
        acc = __builtin_amdgcn_wmma_f32_16x16x4_f32(
            false, a, false, bv, (short)0, acc, false, false);
    }

    // D: VGPR r -> c_local = r + (lane>=16 ? 8 : 0), m = l15
    const int cbase = (lane >> 4) << 3;
    const int m = l15;
    float psum = 0.0f;
#pragma unroll
    for (int r = 0; r < 8; ++r) {
        if (m < M_DIM) {
            const int c = c0 + r + cbase;
            const float p      = acc[r] * (1.0f / (float)HW_DIM); // mean
            const float pooled = cbrtf(p);                        // ^(1/3), p>=0
            const float val    = sqrtf(pooled + 1e-12f);          // sign-sqrt
            out[(size_t)b * MC_DIM + (size_t)m * C_DIM + c] = val;
            psum += pooled + 1e-12f;                              // val²
        }
    }
#pragma unroll
    for (int off = 16; off > 0; off >>= 1)
        psum += __shfl_down(psum, off, 32);
    if (lane == 0) partial[b * CTILES + ctile] = psum;
}

// ---------------------------------------------------------------------------
// Kernel 3: per-batch L2 normalize. Reduce 128 partials, scale 16384 values.
// ---------------------------------------------------------------------------
__global__ __launch_bounds__(256)
void normalize_kernel(const float* __restrict__ partial, float* __restrict__ out) {
    __shared__ float sdata[256];
    const int b   = blockIdx.x;
    const int tid = threadIdx.x;
    sdata[tid] = (tid < CTILES) ? partial[b * CTILES + tid] : 0.0f;
    __syncthreads();
    for (int s = 128; s > 0; s >>= 1) {
        if (tid < s) sdata[tid] += sdata[tid + s];
        __syncthreads();
    }
    const float scale = 1.0f / fmaxf(sqrtf(sdata[0]), 1e-12f);
    float* row = out + (size_t)b * MC_DIM;
    for (int i = tid; i < MC_DIM; i += 256) row[i] *= scale;
}

extern "C" void kernel_launch(void* const* d_in, const int* in_sizes, int n_in,
                              void* d_out, int out_size, void* d_ws, size_t ws_size,
                              hipStream_t stream) {
    (void)in_sizes; (void)n_in; (void)out_size; (void)ws_size;
    const float* F   = (const float*)d_in[0];
    const float* W   = (const float*)d_in[1];
    const float* bnS = (const float*)d_in[2];
    const float* bnB = (const float*)d_in[3];
    const float* bnM = (const float*)d_in[4];
    const float* bnV = (const float*)d_in[5];
    float* out = (float*)d_out;

    float* A3      = (float*)d_ws;                      // 32*16*288 f32 = 576 KiB
    float* partial = A3 + (size_t)B_DIM * 16 * HW_DIM;  // 32*128 f32   =  16 KiB
    float* Wpad    = partial + B_DIM * CTILES;          // 16*2048 f32  = 128 KiB

    pad_w_kernel<<<(16 * C_DIM) / 256, 256, 0, stream>>>(W, Wpad);

    dim3 g1(HW_DIM / 16, B_DIM);   // 18 x 32 blocks, 4 waves each (K-split)
    attn_cube_kernel<<<g1, 32 * K1_WAVES, 0, stream>>>(F, Wpad, bnS, bnB, bnM, bnV, A3);

    dim3 g2(CTILES / 8, B_DIM);    // 16 x 32 blocks, 8 waves each
    gem_pool_kernel<<<g2, 256, 0, stream>>>(F, A3, out, partial);

    normalize_kernel<<<B_DIM, 256, 0, stream>>>(partial, out);
}